// InteractionLayer_78924319031909
// MI455X (gfx1250) — compile-verified
//
#include <hip/hip_runtime.h>

// PhysNet InteractionLayer for gfx1250 (MI455X), fp32 WMMA path.
// F = 128, K = 64 compile-time; N, E from in_sizes.

#define FDIM 128
#define KDIM 64

typedef __attribute__((ext_vector_type(2))) float v2f;
typedef __attribute__((ext_vector_type(8))) float v8f;

__device__ __forceinline__ float ssp(float v) {
    // shifted softplus: log(1+exp(v)) - ln2, numerically stable
    float r = fmaxf(v, 0.0f) + log1pf(__expf(-fabsf(v)));
    return r - 0.69314718055994530942f;
}

// Stage W[ROWS,128] into LDS as interleaved row-pairs:
//   sW2[(k>>1)*128 + col] = { W[k][col], W[k+1][col] }   (k even)
// so a B-fragment is a single aligned ds_load_b64.
template <int ROWS>
__device__ __forceinline__ void stage_w_pairs(const float* __restrict__ W,
                                              v2f* __restrict__ sW2, int tid) {
    constexpr int WORK = (ROWS / 2) * (FDIM / 4);   // row-pair x 4-col groups
    for (int i = tid; i < WORK; i += 256) {
        const int rp = i / (FDIM / 4);
        const int c4 = (i % (FDIM / 4)) * 4;
        const float4 r0 = *(const float4*)(W + (size_t)(2 * rp + 0) * FDIM + c4);
        const float4 r1 = *(const float4*)(W + (size_t)(2 * rp + 1) * FDIM + c4);
        v2f* dst = sW2 + rp * FDIM + c4;
        dst[0] = (v2f){r0.x, r1.x};
        dst[1] = (v2f){r0.y, r1.y};
        dst[2] = (v2f){r0.z, r1.z};
        dst[3] = (v2f){r0.w, r1.w};
    }
}

// ---------------------------------------------------------------------------
// Node GEMM: out[N,128] = epilogue( A_eff[N,128] @ W[128,128] + bias )
//   PREACT: A_eff = ssp(A) element-wise on load
//   MODE 0: out = ssp(acc + bias)
//   MODE 1: out = base + acc + bias
//   MODE 2: out = scale[col]*base[row,col] + acc + bias      (final layer)
// One wave owns a 16-row x 128-col strip; 8 waves/block -> 128 rows.
// ---------------------------------------------------------------------------
template <int MODE, int PREACT>
__global__ void __launch_bounds__(256)
gemm_node(const float* __restrict__ A, const float* __restrict__ W,
          const float* __restrict__ bias, const float* __restrict__ base,
          const float* __restrict__ scale, float* __restrict__ out, int N) {
    __shared__ v2f sW2[(FDIM / 2) * FDIM];   // 64 KB, pair-interleaved

    const int tid = threadIdx.x;
    stage_w_pairs<FDIM>(W, sW2, tid);
    __syncthreads();

    const int wave = tid >> 5;
    const int lane = tid & 31;
    const int half = lane >> 4;     // 0: lanes 0-15, 1: lanes 16-31
    const int l16  = lane & 15;
    const int rowBase = blockIdx.x * 128 + wave * 16;
    if (rowBase >= N) return;       // wave-uniform exit

    v8f acc[8] = {};                // 8 col-blocks of 16x16 f32 accumulators

    // Clamp (not predicate) the A row: garbage rows are never stored.
    int r = rowBase + l16;
    r = (r < N) ? r : (N - 1);
    const float* __restrict__ Arow = A + (size_t)r * FDIM;

    for (int k0 = 0; k0 < FDIM; k0 += 4) {
        // A frag 16x4 f32: lane-half splits K {0,1}/{2,3}; one b64 load.
        const int ka = k0 + 2 * half;
        v2f a = *(const v2f*)(Arow + ka);
        if (PREACT) { a.x = ssp(a.x); a.y = ssp(a.y); }

        const v2f* __restrict__ brow = sW2 + (ka >> 1) * FDIM + l16;
        #pragma unroll
        for (int c = 0; c < 8; ++c) {
            const v2f b = brow[c * 16];   // single ds_load_b64
            acc[c] = __builtin_amdgcn_wmma_f32_16x16x4_f32(
                false, a, false, b, (short)0, acc[c], false, false);
        }
    }

    // Epilogue. C/D layout: VGPR j holds row (j + 8*half), col = c*16 + l16.
    #pragma unroll
    for (int c = 0; c < 8; ++c) {
        const int col = c * 16 + l16;
        const float bcol = bias[col];
        #pragma unroll
        for (int j = 0; j < 8; ++j) {
            const int row = rowBase + j + 8 * half;
            if (row < N) {
                const float v = acc[c][j] + bcol;
                float o;
                if (MODE == 0)      o = ssp(v);
                else if (MODE == 1) o = base[(size_t)row * FDIM + col] + v;
                else                o = scale[col] * base[(size_t)row * FDIM + col] + v;
                out[(size_t)row * FDIM + col] = o;
            }
        }
    }
}

// ---------------------------------------------------------------------------
// Edge kernel, fully fused:
//   g = rbf[e,0:64] @ Wk2f[64,128]          (WMMA, Wk2f pair-staged in LDS)
//   msg = g * xj_all[idx_j[e], :]           (gather; xj_all is L2-resident)
//   m[idx_i[e], :] += msg                   (global_atomic_add_f32)
// One wave per 16-edge tile; 8 waves/block.
// ---------------------------------------------------------------------------
__global__ void __launch_bounds__(256)
edge_gate_scatter(const float* __restrict__ rbf, const float* __restrict__ Wk2f,
                  const int* __restrict__ idx_i, const int* __restrict__ idx_j,
                  const float* __restrict__ xj_all, float* __restrict__ m_acc,
                  int E, int Etiles) {
    __shared__ v2f sW2[(KDIM / 2) * FDIM];   // 32 KB, pair-interleaved

    const int tid = threadIdx.x;
    stage_w_pairs<KDIM>(Wk2f, sW2, tid);
    __syncthreads();

    const int wave = tid >> 5;
    const int lane = tid & 31;
    const int half = lane >> 4;
    const int l16  = lane & 15;
    const int tile = blockIdx.x * 8 + wave;
    if (tile >= Etiles) return;     // wave-uniform exit

    const int e0 = tile * 16;
    v8f acc[8] = {};

    int ea = e0 + l16;              // clamp, not predicate
    ea = (ea < E) ? ea : (E - 1);
    const float* __restrict__ Rrow = rbf + (size_t)ea * KDIM;

    for (int k0 = 0; k0 < KDIM; k0 += 4) {
        const int ka = k0 + 2 * half;
        const v2f a = *(const v2f*)(Rrow + ka);         // one b64 load
        const v2f* __restrict__ brow = sW2 + (ka >> 1) * FDIM + l16;
        #pragma unroll
        for (int c = 0; c < 8; ++c) {
            const v2f b = brow[c * 16];                 // one ds_load_b64
            acc[c] = __builtin_amdgcn_wmma_f32_16x16x4_f32(
                false, a, false, b, (short)0, acc[c], false, false);
        }
    }

    // Gate by gathered xj rows, scatter-add into accumulator (seeded with xi).
    #pragma unroll
    for (int j = 0; j < 8; ++j) {
        const int e = e0 + j + 8 * half;
        if (e < E) {
            const int ji = idx_j[e];
            const int ii = idx_i[e];
            const float* __restrict__ xr = xj_all + (size_t)ji * FDIM;
            float* __restrict__ mr = m_acc + (size_t)ii * FDIM;
            #pragma unroll
            for (int c = 0; c < 8; ++c) {
                const int col = c * 16 + l16;
                atomicAdd(&mr[col], acc[c][j] * xr[col]);
            }
        }
    }
}

// ---------------------------------------------------------------------------
extern "C" void kernel_launch(void* const* d_in, const int* in_sizes, int n_in,
                              void* d_out, int out_size, void* d_ws, size_t ws_size,
                              hipStream_t stream) {
    const float* x    = (const float*)d_in[0];
    const float* rbf  = (const float*)d_in[1];
    const float* Wk2f = (const float*)d_in[2];
    const float* Wi   = (const float*)d_in[3];
    const float* bi   = (const float*)d_in[4];
    const float* Wj   = (const float*)d_in[5];
    const float* bj   = (const float*)d_in[6];
    const float* Wr1  = (const float*)d_in[7];
    const float* br1  = (const float*)d_in[8];
    const float* Wr2  = (const float*)d_in[9];
    const float* br2  = (const float*)d_in[10];
    const float* Wf   = (const float*)d_in[11];
    const float* bf   = (const float*)d_in[12];
    const float* u    = (const float*)d_in[13];
    const int* idx_i  = (const int*)d_in[14];
    const int* idx_j  = (const int*)d_in[15];
    float* out        = (float*)d_out;

    const int N = in_sizes[0] / FDIM;
    const int E = in_sizes[14];

    // Workspace: 3 ping-pong [N,128] fp32 buffers (all L2-resident).
    float* B0 = (float*)d_ws;               // xj_all, later m ping-pong
    float* B1 = B0 + (size_t)N * FDIM;      // xi -> m accumulator
    float* B2 = B1 + (size_t)N * FDIM;      // residual temp t

    const dim3 blk(256);
    const int gb = (N + 127) / 128;

    // xj_all = ssp(ssp(x) @ Wj + bj)
    gemm_node<0, 1><<<gb, blk, 0, stream>>>(x, Wj, bj, nullptr, nullptr, B0, N);
    // m := xi = ssp(ssp(x) @ Wi + bi)   (edge kernel accumulates on top)
    gemm_node<0, 1><<<gb, blk, 0, stream>>>(x, Wi, bi, nullptr, nullptr, B1, N);

    // m += segment_sum( (rbf@Wk2f) * xj_all[idx_j], idx_i )
    const int Etiles = (E + 15) / 16;
    edge_gate_scatter<<<(Etiles + 7) / 8, blk, 0, stream>>>(
        rbf, Wk2f, idx_i, idx_j, B0, B1, E, Etiles);

    // 3 residual blocks: t = ssp(ssp(m)@Wr1+br1); m = m + t@Wr2 + br2
    // l = 0: m B1 -> B0
    gemm_node<0, 1><<<gb, blk, 0, stream>>>(B1, Wr1 + 0 * FDIM * FDIM, br1 + 0 * FDIM,
                                            nullptr, nullptr, B2, N);
    gemm_node<1, 0><<<gb, blk, 0, stream>>>(B2, Wr2 + 0 * FDIM * FDIM, br2 + 0 * FDIM,
                                            B1, nullptr, B0, N);
    // l = 1: m B0 -> B1
    gemm_node<0, 1><<<gb, blk, 0, stream>>>(B0, Wr1 + 1 * FDIM * FDIM, br1 + 1 * FDIM,
                                            nullptr, nullptr, B2, N);
    gemm_node<1, 0><<<gb, blk, 0, stream>>>(B2, Wr2 + 1 * FDIM * FDIM, br2 + 1 * FDIM,
                                            B0, nullptr, B1, N);
    // l = 2: m B1 -> B0
    gemm_node<0, 1><<<gb, blk, 0, stream>>>(B1, Wr1 + 2 * FDIM * FDIM, br1 + 2 * FDIM,
                                            nullptr, nullptr, B2, N);
    gemm_node<1, 0><<<gb, blk, 0, stream>>>(B2, Wr2 + 2 * FDIM * FDIM, br2 + 2 * FDIM,
                                            B1, nullptr, B0, N);

    // out = u * x + ssp(m) @ Wf + bf
    gemm_node<2, 1><<<gb, blk, 0, stream>>>(B0, Wf, bf, x, u, out, N);
}